// Model_47313359732959
// MI455X (gfx1250) — compile-verified
//
#include <hip/hip_runtime.h>
#include <math.h>

// Problem constants (match reference setup_inputs)
#define Bn 8
#define Sn 512
#define Hn 768
#define En 16
#define Mn 4
#define Cn 96
#define Pn (En * (En - 1))   // 240 ordered pairs (hh != tt)

typedef __attribute__((ext_vector_type(2))) float v2f;
typedef __attribute__((ext_vector_type(8))) float v8f;

// ---------------------------------------------------------------------------
// Kernel 1: build maskf[b,e,s] and invcnt[b,e] = 1/max(sum_s mask, 1)
// grid = B*E blocks, block = 512 threads (one thread per token s)
// ---------------------------------------------------------------------------
__global__ void mask_count_kernel(const int* __restrict__ starts,
                                  const int* __restrict__ lens,
                                  float* __restrict__ maskf,
                                  float* __restrict__ invcnt) {
    const int be = blockIdx.x;            // b*E + e
    const int s  = threadIdx.x;           // 0..511
    const int* st = starts + be * Mn;
    const int* ln = lens   + be * Mn;
    int m = 0;
#pragma unroll
    for (int i = 0; i < Mn; ++i) {
        const int a = st[i] + 1;
        const int e = st[i] + ln[i];
        m |= (s >= a) & (s <= e);
    }
    const float f = (float)m;
    maskf[be * Sn + s] = f;

    __shared__ float red[Sn];
    red[s] = f;
    __syncthreads();
    for (int off = Sn / 2; off > 0; off >>= 1) {
        if (s < off) red[s] += red[s + off];
        __syncthreads();
    }
    if (s == 0) invcnt[be] = 1.0f / fmaxf(red[0], 1.0f);
}

// ---------------------------------------------------------------------------
// Kernel 2: att_rows[b,e,t] = (1/count[b,e]) * sum_s maskf[b,e,s]*attention[b,s,t]
// One wave (32 lanes) per 16x16 output tile; K=512 via v_wmma_f32_16x16x4_f32.
// grid = B * (S/16) = 8*32 waves, block = 32 (wave32, EXEC all ones for WMMA)
// ---------------------------------------------------------------------------
__global__ void attrows_wmma_kernel(const float* __restrict__ maskf,
                                    const float* __restrict__ attention,
                                    const float* __restrict__ invcnt,
                                    float* __restrict__ att_rows) {
    const int b    = blockIdx.x >> 5;          // / 32 tiles
    const int tile = blockIdx.x & 31;
    const int t0   = tile * 16;
    const int lane = threadIdx.x;              // 0..31
    const int half = lane >> 4;                // 0: K pair {0,1}; 1: K pair {2,3}
    const int nrow = lane & 15;                // M for A-loads, N for B-loads

    const float* A  = maskf     + b * (En * Sn);   // 16 x 512 row-major
    const float* Bm = attention + b * (Sn * Sn);   // 512 x 512 row-major

    // A layout (32-bit A 16x4): v0 = A[M=nrow][k0+2*half], v1 = A[M=nrow][k0+2*half+1]
    const float* aptr = A + nrow * Sn + 2 * half;
    // B layout (4x16): v0 = B[K=k0+2*half][N=nrow], v1 = B[K=k0+2*half+1][N=nrow]
    const float* bptr = Bm + (size_t)(2 * half) * Sn + t0 + nrow;

    v8f acc = {};
#pragma unroll 4
    for (int k = 0; k < Sn; k += 4) {
        v2f a, bb;
        a.x  = aptr[k + 0];
        a.y  = aptr[k + 1];
        bb.x = bptr[(size_t)k * Sn];
        bb.y = bptr[(size_t)k * Sn + Sn];
        acc = __builtin_amdgcn_wmma_f32_16x16x4_f32(
            /*neg_a=*/false, a, /*neg_b=*/false, bb,
            /*c_mod=*/(short)0, acc, /*reuse_a=*/false, /*reuse_b=*/false);
    }

    // C/D layout: VGPR r -> M = r + 8*half, N = nrow. Fold in 1/count here.
#pragma unroll
    for (int r = 0; r < 8; ++r) {
        const int e = r + 8 * half;
        att_rows[(b * En + e) * Sn + t0 + nrow] = acc[r] * invcnt[b * En + e];
    }
}

// ---------------------------------------------------------------------------
// Kernel 3: context[b,e,h] = sum_t att_rows[b,e,t] * embedding[b,t,h]
// grid = B * (H/16) = 8*48 waves, block = 32
// ---------------------------------------------------------------------------
__global__ void context_wmma_kernel(const float* __restrict__ att_rows,
                                    const float* __restrict__ emb,
                                    float* __restrict__ context) {
    const int b    = blockIdx.x / 48;
    const int tile = blockIdx.x % 48;
    const int h0   = tile * 16;
    const int lane = threadIdx.x;
    const int half = lane >> 4;
    const int nrow = lane & 15;

    const float* A  = att_rows + b * (En * Sn);   // 16 x 512
    const float* Bm = emb      + b * (Sn * Hn);   // 512 x 768

    const float* aptr = A + nrow * Sn + 2 * half;
    const float* bptr = Bm + (size_t)(2 * half) * Hn + h0 + nrow;

    v8f acc = {};
#pragma unroll 4
    for (int k = 0; k < Sn; k += 4) {
        v2f a, bb;
        a.x  = aptr[k + 0];
        a.y  = aptr[k + 1];
        bb.x = bptr[(size_t)k * Hn];
        bb.y = bptr[(size_t)k * Hn + Hn];
        acc = __builtin_amdgcn_wmma_f32_16x16x4_f32(
            false, a, false, bb, (short)0, acc, false, false);
    }

#pragma unroll
    for (int r = 0; r < 8; ++r) {
        const int e = r + 8 * half;
        context[(b * En + e) * Hn + h0 + nrow] = acc[r];
    }
}

// ---------------------------------------------------------------------------
// Kernel 4: doc[b,e,h] = logsumexp over masked tokens of embedding[b,s,h]
// grid = B*E blocks, block = 256. Compact masked indices into LDS first
// (typically <= ~20 tokens), then per-h reduce over the compact list.
// ---------------------------------------------------------------------------
__global__ void doc_lse_kernel(const float* __restrict__ maskf,
                               const float* __restrict__ emb,
                               float* __restrict__ doc) {
    const int be = blockIdx.x;
    const int b  = be / En;
    __shared__ int idx[Sn];
    __shared__ int n;
    if (threadIdx.x == 0) n = 0;
    __syncthreads();
    for (int s = threadIdx.x; s < Sn; s += blockDim.x) {
        if (maskf[be * Sn + s] > 0.5f) {
            const int p = atomicAdd(&n, 1);
            idx[p] = s;
        }
    }
    __syncthreads();
    const int nn = n;
    const float* Eb = emb + (size_t)b * Sn * Hn;
    for (int h = threadIdx.x; h < Hn; h += blockDim.x) {
        float mx = -INFINITY;
        for (int i = 0; i < nn; ++i) mx = fmaxf(mx, Eb[(size_t)idx[i] * Hn + h]);
        float sum = 0.0f;
        for (int i = 0; i < nn; ++i) sum += expf(Eb[(size_t)idx[i] * Hn + h] - mx);
        doc[be * Hn + h] = (nn > 0) ? (mx + logf(sum)) : -INFINITY;
    }
}

// ---------------------------------------------------------------------------
// Kernel 5: assemble output = [feature_set (B*P,4,H) ; labels (B*P,C) as float]
// grid = B*P blocks, block = 256
// pair p -> hh = p/15 ; r = p%15 ; tt = r + (r >= hh)
// ---------------------------------------------------------------------------
__global__ void assemble_kernel(const float* __restrict__ emb,
                                const float* __restrict__ doc,
                                const float* __restrict__ context,
                                const int* __restrict__ labels,
                                float* __restrict__ out) {
    const int bp = blockIdx.x;
    const int b  = bp / Pn;
    const int p  = bp % Pn;
    const int hh = p / (En - 1);
    const int r  = p % (En - 1);
    const int tt = r + (r >= hh ? 1 : 0);

    const float* cls = emb + (size_t)b * Sn * Hn;           // row s = 0
    const float* dh  = doc     + (size_t)(b * En + hh) * Hn;
    const float* dt  = doc     + (size_t)(b * En + tt) * Hn;
    const float* ch  = context + (size_t)(b * En + hh) * Hn;
    const float* ct  = context + (size_t)(b * En + tt) * Hn;

    float* f = out + (size_t)bp * 4 * Hn;
    for (int h = threadIdx.x; h < Hn; h += blockDim.x) {
        f[0 * Hn + h] = cls[h];
        f[1 * Hn + h] = dh[h];
        f[2 * Hn + h] = dt[h];
        f[3 * Hn + h] = ch[h] * ct[h];
    }

    const int* lab = labels + (((size_t)b * En + hh) * En + tt) * Cn;
    float* lo = out + (size_t)Bn * Pn * 4 * Hn + (size_t)bp * Cn;
    for (int c = threadIdx.x; c < Cn; c += blockDim.x)
        lo[c] = (float)lab[c];
}

// ---------------------------------------------------------------------------
extern "C" void kernel_launch(void* const* d_in, const int* in_sizes, int n_in,
                              void* d_out, int out_size, void* d_ws, size_t ws_size,
                              hipStream_t stream) {
    const float* emb       = (const float*)d_in[0];   // (B,S,H)
    const float* attention = (const float*)d_in[1];   // (B,S,S)
    const int*   starts    = (const int*)d_in[2];     // (B,E,M)
    const int*   lens      = (const int*)d_in[3];     // (B,E,M)
    const int*   labels    = (const int*)d_in[4];     // (B,E,E,C)
    float* out = (float*)d_out;

    // Workspace layout (floats)
    float* ws       = (float*)d_ws;
    float* maskf    = ws;                               // B*E*S = 65536
    float* invcnt   = maskf    + (size_t)Bn * En * Sn;  // B*E   = 128
    float* att_rows = invcnt   + (size_t)Bn * En;       // B*E*S = 65536
    float* context  = att_rows + (size_t)Bn * En * Sn;  // B*E*H = 98304
    float* doc      = context  + (size_t)Bn * En * Hn;  // B*E*H = 98304

    mask_count_kernel<<<Bn * En, Sn, 0, stream>>>(starts, lens, maskf, invcnt);
    attrows_wmma_kernel<<<Bn * (Sn / 16), 32, 0, stream>>>(maskf, attention, invcnt, att_rows);
    context_wmma_kernel<<<Bn * (Hn / 16), 32, 0, stream>>>(att_rows, emb, context);
    doc_lse_kernel<<<Bn * En, 256, 0, stream>>>(maskf, emb, doc);
    assemble_kernel<<<Bn * Pn, 256, 0, stream>>>(emb, doc, context, labels, out);
}